// RGCNLinkPredictor_74122545594485
// MI455X (gfx1250) — compile-verified
//
#include <hip/hip_runtime.h>
#include <hip/hip_bf16.h>

typedef __attribute__((ext_vector_type(2))) float v2f;
typedef __attribute__((ext_vector_type(8))) float v8f;

#define D_DIM 128   // D_IN == D_HID == 128

// ---------------------------------------------------------------------------
// Utility: zero an int region
// ---------------------------------------------------------------------------
__global__ void zero_ints(int* __restrict__ p, int n) {
    int i = blockIdx.x * blockDim.x + threadIdx.x;
    if (i < n) p[i] = 0;
}

// ---------------------------------------------------------------------------
// Count edges per (tgt, rel) segment and per relation (for bucketing)
// ---------------------------------------------------------------------------
__global__ void count_edges(const int* __restrict__ edge_index,
                            const int* __restrict__ edge_type,
                            int* __restrict__ cnt,        // [N*R]
                            int* __restrict__ relCount,   // [R]
                            int E, int R) {
    int e = blockIdx.x * blockDim.x + threadIdx.x;
    if (e >= E) return;
    int tgt = edge_index[E + e];
    int t   = edge_type[e];
    atomicAdd(&cnt[tgt * R + t], 1);
    atomicAdd(&relCount[t], 1);
}

// ---------------------------------------------------------------------------
// Tiny exclusive scan over R relation counts (single lane)
// ---------------------------------------------------------------------------
__global__ void scan_rel(const int* __restrict__ relCount,
                         int* __restrict__ offsets,   // [R+1]
                         int* __restrict__ cursor,    // [R]
                         int R) {
    if (threadIdx.x == 0 && blockIdx.x == 0) {
        int acc = 0;
        for (int r = 0; r < R; ++r) {
            offsets[r] = acc;
            cursor[r]  = acc;
            acc += relCount[r];
        }
        offsets[R] = acc;
    }
}

// ---------------------------------------------------------------------------
// Counting-sort edges into relation buckets; precompute per-edge norm
// ---------------------------------------------------------------------------
__global__ void fill_edges(const int* __restrict__ edge_index,
                           const int* __restrict__ edge_type,
                           const int* __restrict__ cnt,
                           int* __restrict__ cursor,
                           int* __restrict__ s_src,
                           int* __restrict__ s_tgt,
                           float* __restrict__ s_norm,
                           int E, int R) {
    int e = blockIdx.x * blockDim.x + threadIdx.x;
    if (e >= E) return;
    int src = edge_index[e];
    int tgt = edge_index[E + e];
    int t   = edge_type[e];
    int p = atomicAdd(&cursor[t], 1);
    s_src[p] = src;
    s_tgt[p] = tgt;
    float c = (float)cnt[tgt * R + t];
    s_norm[p] = 1.0f / fmaxf(c, 1.0f);
}

// ---------------------------------------------------------------------------
// WMMA fp32 GEMM: out[N,128] = X[N,128] @ B[128,128] (+ bias)
//
// 256 threads = 8 waves; wave w computes rows [blk*128 + w*16, +16) x all
// 128 cols as 8 v8f accumulator tiles; 256 v_wmma_f32_16x16x4_f32 per wave.
//
// LDS layout of B: paired-K + parity column swizzle.
//   element (k, col) lives at Bs[(k>>1)*256 + sw(col, (k>>1)&1)*2 + (k&1)]
//   sw(col, par) = (col + par*16) & 127
// => a B fragment {B[k][col], B[k+1][col]} is ONE aligned ds_load_b64, and
//    since the two half-waves of a wave always touch opposite pair-row
//    parities, they map to disjoint 32-bank halves (conflict-free).
// ---------------------------------------------------------------------------
__global__ __launch_bounds__(256) void gemm128_wmma(
        const float* __restrict__ X,
        const float* __restrict__ B,      // [128,128] row-major (K x N)
        float* __restrict__ out,
        const float* __restrict__ bias,   // nullptr -> no bias
        int nrows) {
    __shared__ float Bs[D_DIM * D_DIM];   // 64 KB

    // stage B into paired-K swizzled LDS: 4096 units, 16 per thread
    for (int u = threadIdx.x; u < 4096; u += 256) {
        int p = u >> 6;             // pair row 0..63  (k = 2p, 2p+1)
        int c = (u & 63) << 1;      // even col 0..126
        float2 r0 = *(const float2*)(B + (size_t)(2 * p)     * D_DIM + c);
        float2 r1 = *(const float2*)(B + (size_t)(2 * p + 1) * D_DIM + c);
        int scol = (c + ((p & 1) << 4)) & 127;
        float4 w4;
        w4.x = r0.x; w4.y = r1.x;   // col c   : k=2p, k=2p+1
        w4.z = r0.y; w4.w = r1.y;   // col c+1 : k=2p, k=2p+1
        *(float4*)(Bs + p * 256 + scol * 2) = w4;
    }
    __syncthreads();

    const int wave = threadIdx.x >> 5;
    const int lane = threadIdx.x & 31;
    const int m    = lane & 15;     // A row within tile / B col within tile
    const int hw   = lane >> 4;     // half-wave: 0 or 1
    const int kb   = hw << 1;       // K base within 4-step: 0 or 2
    const int row0 = blockIdx.x * 128 + wave * 16;

    int row  = row0 + m;
    int rowc = row < nrows ? row : (nrows - 1);   // clamp: WMMA needs EXEC all-1s
    const float* xr = X + (size_t)rowc * D_DIM + kb;

    // Per-tile LDS float-index base (constant over the K loop):
    // pair row p = 2*ks + hw  (parity == hw), swizzled col block (t+hw)&7
    int bbase[8];
#pragma unroll
    for (int t = 0; t < 8; ++t)
        bbase[t] = hw * 256 + ((((t + hw) & 7) * 16 + m) << 1);

    v8f acc[8] = {};

#pragma unroll
    for (int ks = 0; ks < 32; ++ks) {           // k0 = 4*ks
        // A fragment: {X[row][k0+kb], X[row][k0+kb+1]}
        float2 av = *(const float2*)(xr + ks * 4);
        v2f a; a.x = av.x; a.y = av.y;
        const float* bsrow = Bs + ks * 512;     // pair row 2*ks
#pragma unroll
        for (int t = 0; t < 8; ++t) {
            v2f b = *(const v2f*)(bsrow + bbase[t]);   // one ds_load_b64
            acc[t] = __builtin_amdgcn_wmma_f32_16x16x4_f32(
                false, a, false, b, (short)0, acc[t], false, false);
        }
    }

    // C layout: VGPR v -> M = v (lanes 0-15) or v+8 (lanes 16-31); N = lane&15
    const int rbase = row0 + (hw << 3);
#pragma unroll
    for (int t = 0; t < 8; ++t) {
        int col = t * 16 + m;
        float bb = bias ? bias[col] : 0.0f;
#pragma unroll
        for (int v = 0; v < 8; ++v) {
            int rr = rbase + v;
            if (rr < nrows) out[(size_t)rr * D_DIM + col] = acc[t][v] + bb;
        }
    }
}

// ---------------------------------------------------------------------------
// Scatter one relation's bucket: h[tgt] += norm * Y[src]  (fp32 atomics)
// One wave per edge, 4 floats per lane; grid-stride over the bucket.
// ---------------------------------------------------------------------------
__global__ __launch_bounds__(256) void scatter_rel(
        const int* __restrict__ offsets, int r,
        const int* __restrict__ s_src,
        const int* __restrict__ s_tgt,
        const float* __restrict__ s_norm,
        const float* __restrict__ Y,
        float* __restrict__ h) {
    const int lane = threadIdx.x & 31;
    const int wid  = blockIdx.x * (blockDim.x >> 5) + (threadIdx.x >> 5);
    const int nw   = gridDim.x * (blockDim.x >> 5);
    const int start = offsets[r];
    const int end   = offsets[r + 1];
    for (int i = start + wid; i < end; i += nw) {
        int src = s_src[i];
        int tgt = s_tgt[i];
        float nrm = s_norm[i];
        float4 y = ((const float4*)(Y + (size_t)src * D_DIM))[lane];
        float* hp = h + (size_t)tgt * D_DIM + lane * 4;
        atomicAdd(hp + 0, nrm * y.x);
        atomicAdd(hp + 1, nrm * y.y);
        atomicAdd(hp + 2, nrm * y.z);
        atomicAdd(hp + 3, nrm * y.w);
    }
}

// ---------------------------------------------------------------------------
// DistMult scoring with fused ReLU: score = sum relu(h[head])*rel*relu(h[tail])
// One wave32 per triplet, 4 dims/lane, shuffle reduction.
// ---------------------------------------------------------------------------
__global__ __launch_bounds__(256) void score_triplets(
        const float* __restrict__ h,
        const float* __restrict__ rel_emb,
        const int* __restrict__ head_idx,
        const int* __restrict__ tail_idx,
        const int* __restrict__ rel_idx,
        float* __restrict__ out, int T) {
    const int lane = threadIdx.x & 31;
    const int t = blockIdx.x * (blockDim.x >> 5) + (threadIdx.x >> 5);
    if (t >= T) return;
    int hi = head_idx[t], ti = tail_idx[t], ri = rel_idx[t];
    float4 hh = ((const float4*)(h       + (size_t)hi * D_DIM))[lane];
    float4 ht = ((const float4*)(h       + (size_t)ti * D_DIM))[lane];
    float4 rr = ((const float4*)(rel_emb + (size_t)ri * D_DIM))[lane];
    float s = fmaxf(hh.x, 0.f) * rr.x * fmaxf(ht.x, 0.f)
            + fmaxf(hh.y, 0.f) * rr.y * fmaxf(ht.y, 0.f)
            + fmaxf(hh.z, 0.f) * rr.z * fmaxf(ht.z, 0.f)
            + fmaxf(hh.w, 0.f) * rr.w * fmaxf(ht.w, 0.f);
#pragma unroll
    for (int o = 16; o > 0; o >>= 1) s += __shfl_xor(s, o, 32);
    if (lane == 0) out[t] = s;
}

// ---------------------------------------------------------------------------
// Host launcher
// ---------------------------------------------------------------------------
extern "C" void kernel_launch(void* const* d_in, const int* in_sizes, int n_in,
                              void* d_out, int out_size, void* d_ws, size_t ws_size,
                              hipStream_t stream) {
    const float* x_emb      = (const float*)d_in[0];
    const float* W          = (const float*)d_in[1];
    const float* W_root     = (const float*)d_in[2];
    const float* bias       = (const float*)d_in[3];
    const float* rel_emb    = (const float*)d_in[4];
    const int*   edge_index = (const int*)d_in[5];
    const int*   edge_type  = (const int*)d_in[6];
    const int*   head_idx   = (const int*)d_in[7];
    const int*   tail_idx   = (const int*)d_in[8];
    const int*   rel_idx    = (const int*)d_in[9];
    float* out = (float*)d_out;

    const int N = in_sizes[0] / D_DIM;
    const int R = in_sizes[4] / D_DIM;
    const int E = in_sizes[6];
    const int T = in_sizes[7];

    // -------- workspace layout (256B aligned) --------
    char* ws = (char*)d_ws;
    size_t off = 0;
    auto alloc = [&](size_t bytes) -> void* {
        void* p = ws + off;
        off = (off + bytes + 255) & ~(size_t)255;
        return p;
    };
    int*   cnt      = (int*)  alloc((size_t)N * R * sizeof(int));   // per (node,rel)
    int*   relCount = (int*)  alloc((size_t)R * sizeof(int));
    int*   offsets  = (int*)  alloc((size_t)(R + 1) * sizeof(int));
    int*   cursor   = (int*)  alloc((size_t)R * sizeof(int));
    int*   s_src    = (int*)  alloc((size_t)E * sizeof(int));
    int*   s_tgt    = (int*)  alloc((size_t)E * sizeof(int));
    float* s_norm   = (float*)alloc((size_t)E * sizeof(float));
    float* h        = (float*)alloc((size_t)N * D_DIM * sizeof(float));
    float* Y        = (float*)alloc((size_t)N * D_DIM * sizeof(float));
    (void)ws_size; (void)n_in; (void)out_size;

    // 1) zero counters
    {
        int n = N * R;
        zero_ints<<<(n + 255) / 256, 256, 0, stream>>>(cnt, n);
        zero_ints<<<1, 64, 0, stream>>>(relCount, R);
    }
    // 2) segment + relation histograms
    count_edges<<<(E + 255) / 256, 256, 0, stream>>>(edge_index, edge_type,
                                                     cnt, relCount, E, R);
    // 3) prefix sum over relations
    scan_rel<<<1, 32, 0, stream>>>(relCount, offsets, cursor, R);
    // 4) bucket edges by relation, precompute norms
    fill_edges<<<(E + 255) / 256, 256, 0, stream>>>(edge_index, edge_type, cnt,
                                                    cursor, s_src, s_tgt, s_norm, E, R);
    // 5) h = x @ W_root + bias (plain stores; must precede scatter atomics)
    const int gemm_blocks = (N + 127) / 128;
    gemm128_wmma<<<gemm_blocks, 256, 0, stream>>>(x_emb, W_root, h, bias, N);
    // 6) per relation: Y = x @ W[r]; h[tgt] += norm * Y[src]
    for (int r = 0; r < R; ++r) {
        gemm128_wmma<<<gemm_blocks, 256, 0, stream>>>(
            x_emb, W + (size_t)r * D_DIM * D_DIM, Y, nullptr, N);
        scatter_rel<<<512, 256, 0, stream>>>(offsets, r, s_src, s_tgt, s_norm, Y, h);
    }
    // 7) DistMult scoring with fused ReLU
    score_triplets<<<(T + 7) / 8, 256, 0, stream>>>(h, rel_emb, head_idx,
                                                    tail_idx, rel_idx, out, T);
}